// VectorQuantizer_89335319757415
// MI455X (gfx1250) — compile-verified
//
#include <hip/hip_runtime.h>
#include <float.h>

typedef __attribute__((ext_vector_type(2))) float v2f;
typedef __attribute__((ext_vector_type(8))) float v8f;

#define B_   16
#define C_   256
#define H_   32
#define W_   32
#define K_   8192
#define D_   256
#define HW_  1024          // H*W
#define CHW_ 262144        // C*H*W
#define N_   16384         // B*H*W
#define ROWS_PER_WG 128
#define LDS_STRIDE  260    // 256 + 4 pad: conflict-free ds_load_b64 fragments
#define KTILE 32           // two 16-wide N sub-tiles staged per iteration
#define QUANT_ELEMS 4194304
#define LOSS_OFF    QUANT_ELEMS
#define ENC_OFF     (QUANT_ELEMS + 1)

// ---- pass 0: codebook row norms (one wave32 per row) ----
__global__ __launch_bounds__(256) void vq_cnorm(const float* __restrict__ cb,
                                                float* __restrict__ cnorm) {
  int row  = (blockIdx.x * blockDim.x + threadIdx.x) >> 5;
  int lane = threadIdx.x & 31;
  if (row >= K_) return;
  const float* r = cb + (size_t)row * D_;
  float s = 0.f;
  for (int j = lane; j < D_; j += 32) { float v = r[j]; s += v * v; }
  #pragma unroll
  for (int m = 16; m >= 1; m >>= 1) s += __shfl_xor(s, m, 32);
  if (lane == 0) cnorm[row] = s;
}

__global__ void vq_zero(float* out) { out[LOSS_OFF] = 0.f; }

// ---- main fused kernel: distances (WMMA f32) + argmin + gather + loss ----
__global__ __launch_bounds__(256) void vq_main(const float* __restrict__ x,
                                               const float* __restrict__ cb,
                                               const float* __restrict__ cnorm,
                                               float* __restrict__ out) {
  __shared__ float bs[KTILE * LDS_STRIDE];  // 32-row codebook tile, shared by 8 waves
  __shared__ int   bestIdx[ROWS_PER_WG];
  __shared__ float lossAcc;

  const int t   = threadIdx.x;
  const int wg  = blockIdx.x;
  const int n0g = wg * ROWS_PER_WG;       // first flat row of this block
  const int b   = n0g >> 10;              // 128 | 1024 -> single batch per block
  const int hw0 = n0g & (HW_ - 1);

  const int wave = t >> 5;
  const int lane = t & 31;
  const int col  = lane & 15;             // N-col within 16-wide sub-tile
  const int hi   = lane >> 4;             // selects K sub-pair per WMMA layout
  const int m0   = wave * 16;             // this wave's local row-tile base

  if (t == 0) lossAcc = 0.f;

  // A matrix (16x4 f32 frag per step) held fully in registers:
  // lane owns x-row (m0+col); elements {d0+2*hi, d0+2*hi+1} for d0 = 0,4,...,252
  const size_t xbase = (size_t)b * CHW_ + (size_t)(hw0 + m0 + col);
  v2f areg[64];
  #pragma unroll
  for (int i = 0; i < 64; ++i) {
    const int c = i * 4 + hi * 2;
    areg[i].x = x[xbase + (size_t)c       * HW_];
    areg[i].y = x[xbase + (size_t)(c + 1) * HW_];
  }

  float minv0[8], minv1[8]; int mini0[8], mini1[8];
  #pragma unroll
  for (int r = 0; r < 8; ++r) {
    minv0[r] = FLT_MAX; mini0[r] = 0;
    minv1[r] = FLT_MAX; mini1[r] = 0;
  }

  for (int kt = 0; kt < K_ / KTILE; ++kt) {
    const int nb = kt * KTILE;
    // cooperative, fully-coalesced stage of 32x256 codebook tile into LDS
    const float4* src = (const float4*)(cb + (size_t)nb * D_);
    #pragma unroll
    for (int j = 0; j < 8; ++j) {
      const int e = j * 256 + t;                 // float4 index in tile (0..2047)
      float4 v = src[e];
      *(float4*)&bs[(e >> 6) * LDS_STRIDE + ((e & 63) << 2)] = v;
    }
    __syncthreads();

    // two independent accumulator chains -> 2x ILP on the matrix pipe,
    // A fragments (registers) reused across both N sub-tiles
    v8f acc0 = {};
    v8f acc1 = {};
    const float* bBase0 = &bs[col        * LDS_STRIDE + hi * 2];
    const float* bBase1 = &bs[(16 + col) * LDS_STRIDE + hi * 2];
    #pragma unroll
    for (int i = 0; i < 64; ++i) {
      v2f b0 = *(const v2f*)(bBase0 + i * 4);
      v2f b1 = *(const v2f*)(bBase1 + i * 4);
      acc0 = __builtin_amdgcn_wmma_f32_16x16x4_f32(
          false, areg[i], false, b0, (short)0, acc0, false, false);
      acc1 = __builtin_amdgcn_wmma_f32_16x16x4_f32(
          false, areg[i], false, b1, (short)0, acc1, false, false);
    }

    const float cn0 = cnorm[nb + col];
    const float cn1 = cnorm[nb + 16 + col];
    #pragma unroll
    for (int r = 0; r < 8; ++r) {
      const float d0 = cn0 - 2.0f * acc0[r];     // ||x||^2 constant per row: dropped
      const bool l0 = d0 < minv0[r];
      minv0[r] = l0 ? d0 : minv0[r];
      mini0[r] = l0 ? (nb + col) : mini0[r];
      const float d1 = cn1 - 2.0f * acc1[r];
      const bool l1 = d1 < minv1[r];
      minv1[r] = l1 ? d1 : minv1[r];
      mini1[r] = l1 ? (nb + 16 + col) : mini1[r];
    }
    __syncthreads();                             // bs reuse guard
  }

  // per-lane merge of the two candidate streams, then cross-lane argmin
  // within each 16-lane half (C/D layout: M = r + 8*hi, N = col)
  #pragma unroll
  for (int r = 0; r < 8; ++r) {
    float v = minv0[r]; int i = mini0[r];
    if (minv1[r] < v || (minv1[r] == v && mini1[r] < i)) { v = minv1[r]; i = mini1[r]; }
    #pragma unroll
    for (int m = 1; m <= 8; m <<= 1) {
      float ov = __shfl_xor(v, m, 32);
      int   oi = __shfl_xor(i, m, 32);
      if (ov < v || (ov == v && oi < i)) { v = ov; i = oi; }
    }
    if (col == 0) bestIdx[m0 + hi * 8 + r] = i;
  }
  __syncthreads();

  // gather quantized rows, transposed coalesced store, fused loss
  {
    const int lrow   = t & 127;
    const int cstart = t >> 7;
    const int idx    = bestIdx[lrow];
    const float* crow = cb + (size_t)idx * D_;
    const size_t base = (size_t)b * CHW_ + (size_t)(hw0 + lrow);
    float a2 = 0.f;
    for (int c = cstart; c < C_; c += 2) {
      const float q  = crow[c];
      const float xv = x[base + (size_t)c * HW_];
      const float df = q - xv;
      a2 += df * df;
      out[base + (size_t)c * HW_] = q;           // straight-through == quant
    }
    #pragma unroll
    for (int m = 16; m >= 1; m >>= 1) a2 += __shfl_xor(a2, m, 32);
    if (lane == 0) atomicAdd(&lossAcc, a2);
  }
  if (t < ROWS_PER_WG) out[ENC_OFF + n0g + t] = (float)bestIdx[t];
  __syncthreads();
  // vq_loss = q_loss + 0.25*e_loss = 1.25 * mean((q - x)^2)
  if (t == 0) atomicAdd(&out[LOSS_OFF], lossAcc * (1.25f / (float)QUANT_ELEMS));
}

extern "C" void kernel_launch(void* const* d_in, const int* in_sizes, int n_in,
                              void* d_out, int out_size, void* d_ws, size_t ws_size,
                              hipStream_t stream) {
  const float* x     = (const float*)d_in[0];
  const float* cb    = (const float*)d_in[1];
  float*       out   = (float*)d_out;
  float*       cnorm = (float*)d_ws;            // 8192 floats of scratch

  vq_zero <<<1, 1, 0, stream>>>(out);
  vq_cnorm<<<K_ / 8, 256, 0, stream>>>(cb, cnorm);     // 8 rows per 256-thread block
  vq_main <<<N_ / ROWS_PER_WG, 256, 0, stream>>>(x, cb, cnorm, out);
}